// ChamferLossLayer_1125281431733
// MI455X (gfx1250) — compile-verified
//
#include <hip/hip_runtime.h>
#include <hip/hip_bf16.h>

typedef float v2f __attribute__((ext_vector_type(2)));
typedef float v8f __attribute__((ext_vector_type(8)));

#define N_BATCH 2
#define NP1 12000
#define NP2 12000

// Pack (x,y,z) -> (x,y,z, x^2+y^2+z^2) as float4, for N*P points.
__global__ void chamfer_prep(const float* __restrict__ cloud,
                             float4* __restrict__ out, int total) {
    int i = blockIdx.x * blockDim.x + threadIdx.x;
    if (i >= total) return;
    float x = cloud[i * 3 + 0];
    float y = cloud[i * 3 + 1];
    float z = cloud[i * 3 + 2];
    out[i] = make_float4(x, y, z, x * x + y * y + z * z);
}

// One wave per 16-row tile. For each 16-col tile of the opposite cloud,
// one V_WMMA_F32_16X16X4_F32 computes the 16x16 block of
// (-2 a.b + ||b||^2); running row-mins kept in 8 VGPRs (C/D layout).
// ||a||^2 is row-constant so it is added after the min loop.
__global__ void __launch_bounds__(32)
chamfer_pass(const float4* __restrict__ rowsP, const float4* __restrict__ colsP,
             int Prows, int Pcols, float* __restrict__ partial) {
    const int lane  = threadIdx.x & 31;
    const int tile  = blockIdx.x;      // row tile index
    const int batch = blockIdx.y;
    const int nTiles = Prows >> 4;

    const float4* rp = rowsP + (size_t)batch * Prows;
    const float2* cp = reinterpret_cast<const float2*>(colsP + (size_t)batch * Pcols);

    const int half = lane >> 4;        // 0: K=0,1   1: K=2,3
    const int l16  = lane & 15;
    const int row  = (tile << 4) + l16;

    // A-matrix 16x4 layout: lane l16 holds row; half selects K pair.
    // A = (-2x, -2y, -2z, 1)
    float2 apair = reinterpret_cast<const float2*>(rp + row)[half];
    v2f a;
    a[0] = -2.0f * apair.x;                      // -2x  or -2z
    a[1] = half ? 1.0f : (-2.0f * apair.y);      // -2y  or  1 (replaces sq)

    const float INF = __builtin_huge_valf();
    float rmin[8];
#pragma unroll
    for (int r = 0; r < 8; ++r) rmin[r] = INF;

    const int jTiles = Pcols >> 4;
#pragma unroll 2
    for (int jt = 0; jt < jTiles; ++jt) {
        // B-matrix 4x16 layout: lane l16 holds column; half selects K pair.
        // B = (x, y, z, sq)  -> raw load from prep array.
        int col = (jt << 4) + l16;
        float2 bp = cp[col * 2 + half];
        v2f b;
        b[0] = bp.x;
        b[1] = bp.y;
        v8f c = {};  // inline zero accumulator
        // D[m][n] = -2 a_m . b_n + ||b_n||^2    (16x16 tile, one WMMA)
        v8f d = __builtin_amdgcn_wmma_f32_16x16x4_f32(
            /*neg_a=*/false, a, /*neg_b=*/false, b,
            /*c_mod=*/(short)0, c, /*reuse_a=*/false, /*reuse_b=*/false);
#pragma unroll
        for (int r = 0; r < 8; ++r) rmin[r] = fminf(rmin[r], d[r]);
    }

    // Cross-lane min within each 16-lane half (xor 1..8 stays in-half).
    // After this, each lane in a half holds rowmin for rows half*8 + r.
#pragma unroll
    for (int off = 8; off >= 1; off >>= 1) {
#pragma unroll
        for (int r = 0; r < 8; ++r)
            rmin[r] = fminf(rmin[r], __shfl_xor(rmin[r], off, 32));
    }

    // Representatives (lanes 0 and 16) add ||a||^2 and sum their 8 rows.
    float s = 0.0f;
    if (l16 == 0) {
#pragma unroll
        for (int r = 0; r < 8; ++r) {
            int m = (tile << 4) + (half << 3) + r;
            s += rmin[r] + rp[m].w;   // + ||a_m||^2
        }
    }
    s += __shfl_xor(s, 16, 32);        // combine the two halves
    if (lane == 0) partial[(size_t)batch * nTiles + tile] = s;
}

// Fixed-order serial reduction per batch -> deterministic output.
__global__ void chamfer_final(const float* __restrict__ part1,
                              const float* __restrict__ part2,
                              float* __restrict__ out, int t1, int t2) {
    int n = threadIdx.x;
    if (n >= N_BATCH) return;
    float s1 = 0.0f, s2 = 0.0f;
    for (int i = 0; i < t1; ++i) s1 += part1[(size_t)n * t1 + i];
    for (int i = 0; i < t2; ++i) s2 += part2[(size_t)n * t2 + i];
    out[n] = s1 / (float)NP1 + s2 / (float)NP2;
}

extern "C" void kernel_launch(void* const* d_in, const int* in_sizes, int n_in,
                              void* d_out, int out_size, void* d_ws, size_t ws_size,
                              hipStream_t stream) {
    const float* c1 = (const float*)d_in[0];   // (N, P1, 3) f32
    const float* c2 = (const float*)d_in[1];   // (N, P2, 3) f32
    float* out = (float*)d_out;                // (N,) f32

    // Workspace layout (floats):
    //   Pa:    N*P1*4   packed cloud1 (x,y,z,sq)
    //   Pb:    N*P2*4   packed cloud2
    //   part1: N*(P1/16)  row-min sums, rows = cloud1
    //   part2: N*(P2/16)  row-min sums, rows = cloud2
    float* Pa    = (float*)d_ws;
    float* Pb    = Pa + (size_t)N_BATCH * NP1 * 4;
    float* part1 = Pb + (size_t)N_BATCH * NP2 * 4;
    float* part2 = part1 + (size_t)N_BATCH * (NP1 / 16);

    const int tot1 = N_BATCH * NP1;
    const int tot2 = N_BATCH * NP2;
    chamfer_prep<<<(tot1 + 255) / 256, 256, 0, stream>>>(c1, (float4*)Pa, tot1);
    chamfer_prep<<<(tot2 + 255) / 256, 256, 0, stream>>>(c2, (float4*)Pb, tot2);

    dim3 g1(NP1 / 16, N_BATCH);   // 750 x 2 single-wave workgroups
    dim3 g2(NP2 / 16, N_BATCH);
    chamfer_pass<<<g1, 32, 0, stream>>>((const float4*)Pa, (const float4*)Pb,
                                        NP1, NP2, part1);
    chamfer_pass<<<g2, 32, 0, stream>>>((const float4*)Pb, (const float4*)Pa,
                                        NP2, NP1, part2);

    chamfer_final<<<1, 32, 0, stream>>>(part1, part2, out, NP1 / 16, NP2 / 16);
}